// DecoderRNN_50594714747315
// MI455X (gfx1250) — compile-verified
//
#include <hip/hip_runtime.h>
#include <hip/hip_bf16.h>

typedef __attribute__((ext_vector_type(2))) float v2f;
typedef __attribute__((ext_vector_type(8))) float v8f;

#define V_SZ   32000
#define E_SZ   128
#define H2_SZ  512
#define TX_SZ  400
#define BS_SZ  16
#define T_DEC  64

// ---------------------------------------------------------------------------
// WMMA f32 16x16x4 helper:  c += A(16xK tile slice) * B(Kx16 tile slice)
// A layout: lane L -> row (L&15), K offsets 2*(L>>4) and +1
// B layout: lane L -> col (L&15), K offsets 2*(L>>4) and +1
// C/D     : vgpr r  -> M = r + 8*(L>>4), N = L&15
// ---------------------------------------------------------------------------
__device__ __forceinline__ v8f wmma_acc(v8f c, v2f a, v2f b) {
    return __builtin_amdgcn_wmma_f32_16x16x4_f32(
        /*neg_a=*/false, a, /*neg_b=*/false, b,
        /*c_mod=*/(short)0, c, /*reuse_a=*/false, /*reuse_b=*/false);
}

// ---------------------------------------------------------------------------
// Init: y0 = emb[START] broadcast, h-state from dec_state
// ---------------------------------------------------------------------------
__global__ void init_kernel(const float* __restrict__ emb,
                            const float* __restrict__ dec_state,
                            float* __restrict__ y,
                            float* __restrict__ h0,
                            float* __restrict__ h1) {
    int tid = threadIdx.x;
    for (int i = tid; i < BS_SZ * E_SZ; i += 256)
        y[i] = emb[1 * E_SZ + (i & (E_SZ - 1))];          // START token = 1
    for (int i = tid; i < BS_SZ * H2_SZ; i += 256) {
        h0[i] = dec_state[i];
        h1[i] = dec_state[BS_SZ * H2_SZ + i];
    }
}

// ---------------------------------------------------------------------------
// Fused GRU cell: grid 32 blocks (one 16-wide column tile of h), 6 waves.
// Waves 0..2: gi gate tiles (r,z,n) = X @ W_ih^T ; waves 3..5: gh tiles.
// Wave-derived control is readfirstlane'd so the K loop is scalar
// (guarantees EXEC==all-1s around every WMMA, as the ISA requires).
// ---------------------------------------------------------------------------
__global__ void gru_kernel(const float* __restrict__ X, int Kx,
                           const float* __restrict__ Hold,
                           const float* __restrict__ Wih,
                           const float* __restrict__ Whh,
                           const float* __restrict__ bih,
                           const float* __restrict__ bhh,
                           float* __restrict__ Hnew) {
    __shared__ float tiles[6][16][16];
    const int wave = __builtin_amdgcn_readfirstlane(threadIdx.x >> 5); // 0..5
    const int lane = threadIdx.x & 31;
    const int j0   = blockIdx.x * 16;           // output column tile
    const int nn   = lane & 15;
    const int khalf = (lane >> 4) << 1;         // 0 or 2
    const int mm   = (lane >> 4) << 3;          // 0 or 8

    const int  gate  = wave % 3;
    const bool hpath = wave >= 3;
    const float* A   = hpath ? Hold : X;
    const int    lda = hpath ? H2_SZ : Kx;
    const int    K   = __builtin_amdgcn_readfirstlane(hpath ? H2_SZ : Kx);
    const float* W   = hpath ? Whh : Wih;
    const int    ldw = K;
    const int    col0 = gate * H2_SZ + j0;

    // dual accumulator chains to break WMMA RAW latency
    v8f c0 = {0.f, 0.f, 0.f, 0.f, 0.f, 0.f, 0.f, 0.f};
    v8f c1 = {0.f, 0.f, 0.f, 0.f, 0.f, 0.f, 0.f, 0.f};
    const float* Arow = A + nn * lda + khalf;
    const float* Wrow = W + (col0 + nn) * ldw + khalf;
    for (int k0 = 0; k0 < K; k0 += 8) {
        v2f a0 = *(const v2f*)(Arow + k0);
        v2f b0 = *(const v2f*)(Wrow + k0);
        v2f a1 = *(const v2f*)(Arow + k0 + 4);
        v2f b1 = *(const v2f*)(Wrow + k0 + 4);
        c0 = wmma_acc(c0, a0, b0);
        c1 = wmma_acc(c1, a1, b1);
    }
    v8f c = c0 + c1;
#pragma unroll
    for (int r = 0; r < 8; ++r) tiles[wave][mm + r][nn] = c[r];
    __syncthreads();

    for (int idx = threadIdx.x; idx < 256; idx += 192) {
        int m = idx >> 4, n = idx & 15;
        int j = j0 + n;
        float ir = tiles[0][m][n] + bih[j];
        float iz = tiles[1][m][n] + bih[H2_SZ + j];
        float in_= tiles[2][m][n] + bih[2 * H2_SZ + j];
        float hr = tiles[3][m][n] + bhh[j];
        float hz = tiles[4][m][n] + bhh[H2_SZ + j];
        float hn = tiles[5][m][n] + bhh[2 * H2_SZ + j];
        float rg = 1.f / (1.f + expf(-(ir + hr)));
        float zg = 1.f / (1.f + expf(-(iz + hz)));
        float ng = tanhf(in_ + rg * hn);
        float ho = Hold[m * H2_SZ + j];
        Hnew[m * H2_SZ + j] = (1.f - zg) * ng + zg * ho;
    }
}

// ---------------------------------------------------------------------------
// q = dec_out @ Wa   (M=16, N=512, K=512, B is K-major). grid 4 x 256.
// ---------------------------------------------------------------------------
__global__ void q_kernel(const float* __restrict__ dec,
                         const float* __restrict__ Wa,
                         float* __restrict__ q) {
    const int wave = __builtin_amdgcn_readfirstlane(threadIdx.x >> 5);
    const int lane = threadIdx.x & 31;
    const int j0 = (blockIdx.x * 8 + wave) * 16;
    const int nn = lane & 15;
    const int khalf = (lane >> 4) << 1;
    const int mm = (lane >> 4) << 3;

    v8f c0 = {0.f, 0.f, 0.f, 0.f, 0.f, 0.f, 0.f, 0.f};
    v8f c1 = {0.f, 0.f, 0.f, 0.f, 0.f, 0.f, 0.f, 0.f};
    const float* Arow = dec + nn * H2_SZ + khalf;
    const float* Bcol = Wa + khalf * H2_SZ + j0 + nn;
    for (int k0 = 0; k0 < H2_SZ; k0 += 8) {
        v2f a0 = *(const v2f*)(Arow + k0);
        v2f a1 = *(const v2f*)(Arow + k0 + 4);
        v2f b0, b1;
        b0.x = Bcol[(k0    ) * H2_SZ];
        b0.y = Bcol[(k0 + 1) * H2_SZ];
        b1.x = Bcol[(k0 + 4) * H2_SZ];
        b1.y = Bcol[(k0 + 5) * H2_SZ];
        c0 = wmma_acc(c0, a0, b0);
        c1 = wmma_acc(c1, a1, b1);
    }
    v8f c = c0 + c1;
#pragma unroll
    for (int r = 0; r < 8; ++r) q[(mm + r) * H2_SZ + j0 + nn] = c[r];
}

// ---------------------------------------------------------------------------
// scores[b,t] = q[b,:] . h_j[b,t,:]   grid (50,16) x 256 (wave per t).
// ---------------------------------------------------------------------------
__global__ void scores_kernel(const float* __restrict__ q,
                              const float* __restrict__ hj,
                              float* __restrict__ sc) {
    const int wave = threadIdx.x >> 5;
    const int lane = threadIdx.x & 31;
    const int t = blockIdx.x * 8 + wave;
    const int b = blockIdx.y;
    const float* qr = q + b * H2_SZ;
    const float* hr = hj + (b * TX_SZ + t) * H2_SZ;
    float s = 0.f;
    for (int k = lane; k < H2_SZ; k += 32) s += qr[k] * hr[k];
    for (int off = 16; off > 0; off >>= 1) s += __shfl_xor(s, off, 32);
    if (lane == 0) sc[b * TX_SZ + t] = s;
}

// ---------------------------------------------------------------------------
// softmax over Tx=400, writes a_ij directly to the weights output region.
// grid 16 x 256.
// ---------------------------------------------------------------------------
__global__ void softmax_kernel(const float* __restrict__ sc,
                               float* __restrict__ wout, int tstep) {
    __shared__ float red[256];
    const int b = blockIdx.x, tid = threadIdx.x;
    const float* row = sc + b * TX_SZ;
    float m = -3.4e38f;
    for (int i = tid; i < TX_SZ; i += 256) m = fmaxf(m, row[i]);
    red[tid] = m; __syncthreads();
    for (int off = 128; off > 0; off >>= 1) {
        if (tid < off) red[tid] = fmaxf(red[tid], red[tid + off]);
        __syncthreads();
    }
    const float bm = red[0]; __syncthreads();
    float s = 0.f;
    for (int i = tid; i < TX_SZ; i += 256) s += expf(row[i] - bm);
    red[tid] = s; __syncthreads();
    for (int off = 128; off > 0; off >>= 1) {
        if (tid < off) red[tid] += red[tid + off];
        __syncthreads();
    }
    const float inv = 1.f / red[0];
    float* w = wout + b * (T_DEC * TX_SZ) + tstep * TX_SZ;
    for (int i = tid; i < TX_SZ; i += 256) w[i] = expf(row[i] - bm) * inv;
}

// ---------------------------------------------------------------------------
// ctx[b,h] = sum_t a[b,t] * h_j[b,t,h]   grid (2,16) x 256.
// ---------------------------------------------------------------------------
__global__ void ctx_kernel(const float* __restrict__ w,
                           const float* __restrict__ hj,
                           float* __restrict__ ctx, int tstep) {
    const int b = blockIdx.y;
    const int h = blockIdx.x * 256 + threadIdx.x;
    const float* wr = w + b * (T_DEC * TX_SZ) + tstep * TX_SZ;
    const float* hb = hj + b * TX_SZ * H2_SZ + h;
    float acc = 0.f;
    for (int t = 0; t < TX_SZ; ++t) acc += wr[t] * hb[t * H2_SZ];
    ctx[b * H2_SZ + h] = acc;
}

// ---------------------------------------------------------------------------
// att = tanh([ctx; dec] @ Wc + bc)   M=16, N=512, K=1024, B K-major. grid 4x256
// ---------------------------------------------------------------------------
__global__ void att_kernel(const float* __restrict__ ctx,
                           const float* __restrict__ dec,
                           const float* __restrict__ Wc,
                           const float* __restrict__ bc,
                           float* __restrict__ att) {
    const int wave = __builtin_amdgcn_readfirstlane(threadIdx.x >> 5);
    const int lane = threadIdx.x & 31;
    const int j0 = (blockIdx.x * 8 + wave) * 16;
    const int nn = lane & 15;
    const int khalf = (lane >> 4) << 1;
    const int mm = (lane >> 4) << 3;

    v8f c0 = {0.f, 0.f, 0.f, 0.f, 0.f, 0.f, 0.f, 0.f};
    v8f c1 = {0.f, 0.f, 0.f, 0.f, 0.f, 0.f, 0.f, 0.f};
    const float* Bcol = Wc + khalf * H2_SZ + j0 + nn;
    for (int k0 = 0; k0 < 2 * H2_SZ; k0 += 8) {
        const float* A = (k0 < H2_SZ) ? ctx : dec;
        const int kk = (k0 < H2_SZ) ? k0 : (k0 - H2_SZ);
        v2f a0 = *(const v2f*)(A + nn * H2_SZ + kk + khalf);
        v2f a1 = *(const v2f*)(A + nn * H2_SZ + kk + khalf + 4);
        v2f b0, b1;
        b0.x = Bcol[(k0    ) * H2_SZ];
        b0.y = Bcol[(k0 + 1) * H2_SZ];
        b1.x = Bcol[(k0 + 4) * H2_SZ];
        b1.y = Bcol[(k0 + 5) * H2_SZ];
        c0 = wmma_acc(c0, a0, b0);
        c1 = wmma_acc(c1, a1, b1);
    }
    v8f c = c0 + c1;
#pragma unroll
    for (int r = 0; r < 8; ++r)
        att[(mm + r) * H2_SZ + j0 + nn] = tanhf(c[r] + bc[j0 + nn]);
}

// ---------------------------------------------------------------------------
// e = att @ Wd2e^T + bd2e   M=16, N=128, K=512, B N-major. grid 1 x 256.
// ---------------------------------------------------------------------------
__global__ void e_kernel(const float* __restrict__ att,
                         const float* __restrict__ Wd2e,
                         const float* __restrict__ bd2e,
                         float* __restrict__ eo) {
    const int wave = __builtin_amdgcn_readfirstlane(threadIdx.x >> 5);
    const int lane = threadIdx.x & 31;
    const int j0 = wave * 16;
    const int nn = lane & 15;
    const int khalf = (lane >> 4) << 1;
    const int mm = (lane >> 4) << 3;

    v8f c0 = {0.f, 0.f, 0.f, 0.f, 0.f, 0.f, 0.f, 0.f};
    v8f c1 = {0.f, 0.f, 0.f, 0.f, 0.f, 0.f, 0.f, 0.f};
    const float* Arow = att + nn * H2_SZ + khalf;
    const float* Wrow = Wd2e + (j0 + nn) * H2_SZ + khalf;
    for (int k0 = 0; k0 < H2_SZ; k0 += 8) {
        v2f a0 = *(const v2f*)(Arow + k0);
        v2f b0 = *(const v2f*)(Wrow + k0);
        v2f a1 = *(const v2f*)(Arow + k0 + 4);
        v2f b1 = *(const v2f*)(Wrow + k0 + 4);
        c0 = wmma_acc(c0, a0, b0);
        c1 = wmma_acc(c1, a1, b1);
    }
    v8f c = c0 + c1;
#pragma unroll
    for (int r = 0; r < 8; ++r)
        eo[(mm + r) * E_SZ + j0 + nn] = c[r] + bd2e[j0 + nn];
}

// ---------------------------------------------------------------------------
// logits = e @ Wout^T + bout   M=16, N=32000, K=128. grid 250 x 256.
// ---------------------------------------------------------------------------
__global__ void logits_kernel(const float* __restrict__ e,
                              const float* __restrict__ Wout,
                              const float* __restrict__ bout,
                              float* __restrict__ logits) {
    const int wave = __builtin_amdgcn_readfirstlane(threadIdx.x >> 5);
    const int lane = threadIdx.x & 31;
    const int j0 = (blockIdx.x * 8 + wave) * 16;
    const int nn = lane & 15;
    const int khalf = (lane >> 4) << 1;
    const int mm = (lane >> 4) << 3;

    v8f c0 = {0.f, 0.f, 0.f, 0.f, 0.f, 0.f, 0.f, 0.f};
    v8f c1 = {0.f, 0.f, 0.f, 0.f, 0.f, 0.f, 0.f, 0.f};
    const float* Arow = e + nn * E_SZ + khalf;
    const float* Wrow = Wout + (j0 + nn) * E_SZ + khalf;
#pragma unroll
    for (int k0 = 0; k0 < E_SZ; k0 += 8) {
        v2f a0 = *(const v2f*)(Arow + k0);
        v2f b0 = *(const v2f*)(Wrow + k0);
        v2f a1 = *(const v2f*)(Arow + k0 + 4);
        v2f b1 = *(const v2f*)(Wrow + k0 + 4);
        c0 = wmma_acc(c0, a0, b0);
        c1 = wmma_acc(c1, a1, b1);
    }
    v8f c = c0 + c1;
#pragma unroll
    for (int r = 0; r < 8; ++r)
        logits[(mm + r) * V_SZ + j0 + nn] = c[r] + bout[j0 + nn];
}

// ---------------------------------------------------------------------------
// per-b max/argmax over V, logsumexp, y_next = emb[argmax]. grid 16 x 256.
// ---------------------------------------------------------------------------
__global__ void reduce_kernel(const float* __restrict__ logits,
                              const float* __restrict__ emb,
                              float* __restrict__ lse,
                              float* __restrict__ ynext) {
    __shared__ float smax[256];
    __shared__ int   sidx[256];
    __shared__ float ssum[256];
    const int b = blockIdx.x, tid = threadIdx.x;
    const float* row = logits + b * V_SZ;
    float m = -3.4e38f; int mi = 0;
    for (int v = tid; v < V_SZ; v += 256) {
        float x = row[v];
        if (x > m) { m = x; mi = v; }
    }
    smax[tid] = m; sidx[tid] = mi; __syncthreads();
    for (int off = 128; off > 0; off >>= 1) {
        if (tid < off) {
            float a = smax[tid], bb = smax[tid + off];
            int ai = sidx[tid], bi = sidx[tid + off];
            if (bb > a || (bb == a && bi < ai)) { smax[tid] = bb; sidx[tid] = bi; }
        }
        __syncthreads();
    }
    const float bm = smax[0];
    const int bidx = sidx[0];
    float s = 0.f;
    for (int v = tid; v < V_SZ; v += 256) s += expf(row[v] - bm);
    ssum[tid] = s; __syncthreads();
    for (int off = 128; off > 0; off >>= 1) {
        if (tid < off) ssum[tid] += ssum[tid + off];
        __syncthreads();
    }
    if (tid == 0) lse[b] = bm + logf(ssum[0]);
    if (tid < E_SZ) ynext[b * E_SZ + tid] = emb[bidx * E_SZ + tid];
}

// ---------------------------------------------------------------------------
// logp writeback into (b, v, t) strided output. grid 2000 x 256.
// ---------------------------------------------------------------------------
__global__ void writeback_kernel(const float* __restrict__ logits,
                                 const float* __restrict__ lse,
                                 float* __restrict__ outp, int tstep) {
    int g = blockIdx.x * 256 + threadIdx.x;          // 0 .. 16*32000-1
    int b = g / V_SZ, v = g - b * V_SZ;
    outp[b * (V_SZ * T_DEC) + v * T_DEC + tstep] = logits[g] - lse[b];
}

// ---------------------------------------------------------------------------
extern "C" void kernel_launch(void* const* d_in, const int* in_sizes, int n_in,
                              void* d_out, int out_size, void* d_ws, size_t ws_size,
                              hipStream_t stream) {
    const float* hj    = (const float*)d_in[0];
    const float* dec0  = (const float*)d_in[1];
    // d_in[2] = ys_prev (unused: eval-mode greedy decode)
    const float* emb   = (const float*)d_in[3];
    const float* Wih0  = (const float*)d_in[4];
    const float* Whh0  = (const float*)d_in[5];
    const float* bih0  = (const float*)d_in[6];
    const float* bhh0  = (const float*)d_in[7];
    const float* Wih1  = (const float*)d_in[8];
    const float* Whh1  = (const float*)d_in[9];
    const float* bih1  = (const float*)d_in[10];
    const float* bhh1  = (const float*)d_in[11];
    const float* Wa    = (const float*)d_in[12];
    const float* Wc    = (const float*)d_in[13];
    const float* bc    = (const float*)d_in[14];
    const float* Wd2e  = (const float*)d_in[15];
    const float* bd2e  = (const float*)d_in[16];
    const float* Wout  = (const float*)d_in[17];
    const float* bout  = (const float*)d_in[18];

    float* outputs = (float*)d_out;                         // (16, 32000, 64)
    float* weights = (float*)d_out + BS_SZ * V_SZ * T_DEC;  // (16, 64, 400)

    float* ws = (float*)d_ws;
    float* logits = ws;                  // 512000
    float* h0A = ws + 512000;            // 8192
    float* h0B = ws + 520192;
    float* h1A = ws + 528384;
    float* h1B = ws + 536576;
    float* yA  = ws + 544768;            // 2048
    float* yB  = ws + 546816;
    float* q   = ws + 548864;            // 8192
    float* sc  = ws + 557056;            // 6400
    float* ctx = ws + 563456;            // 8192
    float* att = ws + 571648;            // 8192
    float* eb  = ws + 579840;            // 2048
    float* lse = ws + 581888;            // 16

    init_kernel<<<1, 256, 0, stream>>>(emb, dec0, yA, h0A, h1A);

    float* h0c = h0A; float* h0n = h0B;
    float* h1c = h1A; float* h1n = h1B;
    float* yc  = yA;  float* yn  = yB;

    for (int t = 0; t < T_DEC; ++t) {
        gru_kernel<<<32, 192, 0, stream>>>(yc, E_SZ, h0c, Wih0, Whh0, bih0, bhh0, h0n);
        gru_kernel<<<32, 192, 0, stream>>>(h0n, H2_SZ, h1c, Wih1, Whh1, bih1, bhh1, h1n);
        q_kernel<<<4, 256, 0, stream>>>(h1n, Wa, q);
        scores_kernel<<<dim3(50, 16), 256, 0, stream>>>(q, hj, sc);
        softmax_kernel<<<16, 256, 0, stream>>>(sc, weights, t);
        ctx_kernel<<<dim3(2, 16), 256, 0, stream>>>(weights, hj, ctx, t);
        att_kernel<<<4, 256, 0, stream>>>(ctx, h1n, Wc, bc, att);
        e_kernel<<<1, 256, 0, stream>>>(att, Wd2e, bd2e, eb);
        logits_kernel<<<250, 256, 0, stream>>>(eb, Wout, bout, logits);
        reduce_kernel<<<16, 256, 0, stream>>>(logits, emb, lse, yn);
        writeback_kernel<<<2000, 256, 0, stream>>>(logits, lse, outputs, t);

        float* tmp;
        tmp = h0c; h0c = h0n; h0n = tmp;
        tmp = h1c; h1c = h1n; h1n = tmp;
        tmp = yc;  yc  = yn;  yn  = tmp;
    }
}